// SGConv_3178275799582
// MI455X (gfx1250) — compile-verified
//
#include <hip/hip_runtime.h>
#include <stdint.h>

#define F_DIM    48
#define TILE     256
#define NTHREADS 256

// Low 32 bits of a flat shared-memory address are the LDS byte offset
// (ISA 10.2 aperture rules).
__device__ __forceinline__ uint32_t lds_addr_of(const void* p) {
  return (uint32_t)(uintptr_t)p;
}

// CDNA5 async DMA: global -> LDS, GVS addressing (SADDR base + 32-bit VGPR
// byte offset). Tracked with ASYNCcnt. CDNA5 ISA.
__device__ __forceinline__ void async_load_b32(uint32_t lds, int goff_bytes,
                                               const void* base) {
  asm volatile("global_load_async_to_lds_b32 %0, %1, %2"
               :
               : "v"(lds), "v"(goff_bytes), "s"(base)
               : "memory");
}

__device__ __forceinline__ void wait_async0() {
  asm volatile("s_wait_asynccnt 0x0" ::: "memory");
}

__device__ __forceinline__ void prefetch_tile(int tile, int E,
                                              const int* esrc, const int* edst,
                                              const float* ew,
                                              int* s_src, int* s_dst, float* s_w,
                                              int tid) {
  const int e = tile * TILE + tid;
  if (e < E) {
    const int gb = e * 4;
    async_load_b32(lds_addr_of(&s_src[tid]), gb, esrc);
    async_load_b32(lds_addr_of(&s_dst[tid]), gb, edst);
    async_load_b32(lds_addr_of(&s_w[tid]),   gb, ew);
  }
}

__global__ void __launch_bounds__(NTHREADS)
spmm_scatter_kernel(const float* __restrict__ x,
                    const float* __restrict__ ew,
                    const int* __restrict__ esrc,
                    const int* __restrict__ edst,
                    float* __restrict__ out,
                    int E, int numTiles) {
  __shared__ int   s_src[2][TILE];
  __shared__ int   s_dst[2][TILE];
  __shared__ float s_w[2][TILE];

  const int tid  = threadIdx.x;
  const int wid  = tid >> 5;   // wave32: 8 waves per block
  const int lane = tid & 31;

  // Edge-pair lane mapping: 2 edges x 48 features = 96 = 3 full wave32 steps.
  // step0: edgeA f = lane      | step1: lanes 0-15 edgeA f=32+lane,
  // step2: edgeB f = 16 + lane |        lanes 16-31 edgeB f=lane-16
  const bool hiHalf = (lane >= 16);
  const int  f1     = hiHalf ? (lane - 16) : (32 + lane);

  int t   = blockIdx.x;
  int buf = 0;

  if (t < numTiles)
    prefetch_tile(t, E, esrc, edst, ew, s_src[0], s_dst[0], s_w[0], tid);

  for (; t < numTiles; t += gridDim.x) {
    wait_async0();        // current tile's DMA complete (this wave's issues)
    __syncthreads();      // all waves' portions visible in LDS

    const int tn = t + (int)gridDim.x;
    if (tn < numTiles)    // overlap next tile's DMA with current compute
      prefetch_tile(tn, E, esrc, edst, ew,
                    s_src[buf ^ 1], s_dst[buf ^ 1], s_w[buf ^ 1], tid);

    const int tile_base = t * TILE;
    const int count = (E - tile_base < TILE) ? (E - tile_base) : TILE;
    const int ebeg = wid * 32;
    int eend = ebeg + 32;
    if (eend > count) eend = count;

    int e = ebeg;
    for (; e + 1 < eend; e += 2) {
      const int   srcA = s_src[buf][e];       // LDS broadcast
      const int   dstA = s_dst[buf][e];
      const float wA   = s_w[buf][e];
      const int   srcB = s_src[buf][e + 1];
      const int   dstB = s_dst[buf][e + 1];
      const float wB   = s_w[buf][e + 1];

      const int   srcM = hiHalf ? srcB : srcA;  // v_cndmask, no divergence
      const int   dstM = hiHalf ? dstB : dstA;
      const float wM   = hiHalf ? wB   : wA;

      const float v0 = wA * x[srcA * F_DIM + lane];
      const float v1 = wM * x[srcM * F_DIM + f1];
      const float v2 = wB * x[srcB * F_DIM + 16 + lane];

      unsafeAtomicAdd(&out[dstA * F_DIM + lane],      v0);  // global_atomic_add_f32
      unsafeAtomicAdd(&out[dstM * F_DIM + f1],        v1);
      unsafeAtomicAdd(&out[dstB * F_DIM + 16 + lane], v2);
    }
    if (e < eend) {       // odd tail edge in this wave's slice
      const int   src = s_src[buf][e];
      const int   dst = s_dst[buf][e];
      const float w   = s_w[buf][e];
      unsafeAtomicAdd(&out[dst * F_DIM + lane], w * x[src * F_DIM + lane]);
      if (lane < 16)
        unsafeAtomicAdd(&out[dst * F_DIM + 32 + lane],
                        w * x[src * F_DIM + 32 + lane]);
    }

    __syncthreads();      // all reads of buf done before it is re-filled
    buf ^= 1;
  }
}

__global__ void zero_kernel(float4* __restrict__ p, int n4) {
  int i = blockIdx.x * blockDim.x + threadIdx.x;
  const int stride = gridDim.x * blockDim.x;
  const float4 z = make_float4(0.f, 0.f, 0.f, 0.f);
  for (; i < n4; i += stride) p[i] = z;   // global_store_b128
}

extern "C" void kernel_launch(void* const* d_in, const int* in_sizes, int n_in,
                              void* d_out, int out_size, void* d_ws, size_t ws_size,
                              hipStream_t stream) {
  const float* x    = (const float*)d_in[0];   // [N, 48] fp32
  const float* ew   = (const float*)d_in[1];   // [E] fp32
  const int*   esrc = (const int*)d_in[2];     // [E] i32
  const int*   edst = (const int*)d_in[3];     // [E] i32
  float*       out  = (float*)d_out;           // [N, 48] fp32
  float*       tmp  = (float*)d_ws;            // intermediate x' (19.2 MB)

  const int NF = in_sizes[0];                  // N * 48, divisible by 4
  const int E  = in_sizes[1];
  const int numTiles = (E + TILE - 1) / TILE;

  zero_kernel<<<1024, 256, 0, stream>>>((float4*)tmp, NF / 4);
  zero_kernel<<<1024, 256, 0, stream>>>((float4*)out, NF / 4);

  int blocks = numTiles < 2048 ? numTiles : 2048;

  // Iteration 1: x -> tmp
  spmm_scatter_kernel<<<blocks, NTHREADS, 0, stream>>>(x, ew, esrc, edst, tmp,
                                                       E, numTiles);
  // Iteration 2: tmp -> out
  spmm_scatter_kernel<<<blocks, NTHREADS, 0, stream>>>(tmp, ew, esrc, edst, out,
                                                       E, numTiles);
}